// hypergraph_enconder_40218073760225
// MI455X (gfx1250) — compile-verified
//
#include <hip/hip_runtime.h>
#include <hip/hip_bf16.h>

typedef __attribute__((ext_vector_type(2))) float v2f;
typedef __attribute__((ext_vector_type(8))) float v8f;

__device__ __forceinline__ v8f wmma_f32_k4(v2f a, v2f b, v8f c) {
    // D = A(16x4,f32) * B(4x16,f32) + C(16x16,f32), full fp32 precision
    return __builtin_amdgcn_wmma_f32_16x16x4_f32(false, a, false, b, (short)0, c, false, false);
}

__device__ __forceinline__ float lrelu(float v) { return v > 0.f ? v : 0.2f * v; }

// ---------------------------------------------------------------- gather
__global__ void gather_kernel(const float* __restrict__ emb, const int* __restrict__ seq,
                              float* __restrict__ X, int N) {
    int idx = blockIdx.x * 256 + threadIdx.x;
    if (idx < N * 64) {
        int row = idx >> 6, col = idx & 63;
        X[idx] = emb[(size_t)seq[row] * 64 + col];
    }
}

// ------------------------------------------------------- column sum (deterministic)
__global__ void colsum_kernel(const float* __restrict__ X, float* __restrict__ colsum, int N) {
    int col = blockIdx.x;          // 0..63
    int t = threadIdx.x;           // 256
    float acc = 0.f;
    for (int r = t; r < N; r += 256) acc += X[(size_t)r * 64 + col];
    __shared__ float red[256];
    red[t] = acc; __syncthreads();
    for (int s = 128; s > 0; s >>= 1) { if (t < s) red[t] += red[t + s]; __syncthreads(); }
    if (t == 0) colsum[col] = red[0];
}

// ------------------------------------- mean row @ W_conv + b_conv (3 modalities)
__global__ void meanrow_kernel(const float* __restrict__ colsum, const float* __restrict__ Wc,
                               const float* __restrict__ bc, float* __restrict__ rows3,
                               int Nd, int Np, int Nm) {
    int mod = blockIdx.x, j = threadIdx.x;      // 64 threads
    int N = (mod == 0) ? Nd : (mod == 1 ? Np : Nm);
    const float* s = colsum + mod * 64;
    float acc = 0.f;
    for (int k = 0; k < 64; ++k) acc += s[k] * Wc[k * 64 + j];
    rows3[mod * 64 + j] = acc / (float)N + bc[j];
}

// ------------------------------ broadcast per-modality rows into out[:,64:128]
__global__ void broadcast_kernel(const float* __restrict__ rows3, float* __restrict__ out,
                                 int Nd, int Np, int Nm) {
    int idx = blockIdx.x * 256 + threadIdx.x;
    int total = (Nd + Np + Nm) * 64;
    if (idx < total) {
        int row = idx >> 6, col = idx & 63;
        int mod = (row < Nd) ? 0 : (row < Nd + Np ? 1 : 2);
        out[(size_t)row * 128 + 64 + col] = rows3[mod * 64 + col];
    }
}

// ------------- C[M x 64] = A[M x 64] @ W[64 x 64]; one wave = 16x64 strip,
// ------------- 4 independent accumulators, A fragment loaded once per K-step
__global__ void gemm_xw_kernel(const float* __restrict__ A, const float* __restrict__ W,
                               float* __restrict__ C) {
    int tileM = blockIdx.x;
    int lane = threadIdx.x;        // 32 threads = 1 wave
    int half = lane >> 4, l16 = lane & 15;
    int rowA = tileM * 16 + l16;
    const float* Arow = A + (size_t)rowA * 64;
    v8f c0 = {}, c1 = {}, c2 = {}, c3 = {};
#pragma unroll
    for (int k0 = 0; k0 < 64; k0 += 4) {
        int ka = k0 + 2 * half;
        v2f a;
        a.x = Arow[ka];
        a.y = Arow[ka + 1];
        const float* w0 = W + (size_t)ka * 64 + l16;
        const float* w1 = w0 + 64;
        v2f b0, b1, b2, b3;
        b0.x = w0[0];  b0.y = w1[0];
        b1.x = w0[16]; b1.y = w1[16];
        b2.x = w0[32]; b2.y = w1[32];
        b3.x = w0[48]; b3.y = w1[48];
        c0 = wmma_f32_k4(a, b0, c0);
        c1 = wmma_f32_k4(a, b1, c1);
        c2 = wmma_f32_k4(a, b2, c2);
        c3 = wmma_f32_k4(a, b3, c3);
    }
#pragma unroll
    for (int r = 0; r < 8; ++r) {
        int row = tileM * 16 + r + 8 * half;
        float* crow = C + (size_t)row * 64 + l16;
        crow[0]  = c0[r];
        crow[16] = c1[r];
        crow[32] = c2[r];
        crow[48] = c3[r];
    }
}

// ------------------------------------------- s[i] = dot(Y[i,:64], a[:64])
__global__ void rowdot_kernel(const float* __restrict__ Y, const float* __restrict__ av,
                              float* __restrict__ s, int M) {
    __shared__ float a[64];
    if (threadIdx.x < 64) a[threadIdx.x] = av[threadIdx.x];
    __syncthreads();
    int i = blockIdx.x * blockDim.x + threadIdx.x;
    if (i < M) {
        float acc = 0.f;
        for (int k = 0; k < 64; ++k) acc += Y[(size_t)i * 64 + k] * a[k];
        s[i] = acc;
    }
}

// ------------------- per-edge softmax over {self} ∪ {medicine nodes} (block per edge)
__global__ void softmax_kernel(const float* __restrict__ sn, const float* __restrict__ se,
                               float* __restrict__ A, float* __restrict__ aself,
                               int Nm, int Nn) {
    int d = blockIdx.x;
    int t = threadIdx.x;           // 256
    float sed = se[d];
    float vself = lrelu(sn[d] + sed);
    float mx = (t == 0) ? vself : -1e30f;
    for (int m = t; m < Nm; m += 256) mx = fmaxf(mx, lrelu(sn[Nn + m] + sed));
    __shared__ float red[256];
    red[t] = mx; __syncthreads();
    for (int s = 128; s > 0; s >>= 1) { if (t < s) red[t] = fmaxf(red[t], red[t + s]); __syncthreads(); }
    mx = red[0]; __syncthreads();
    float sum = (t == 0) ? expf(vself - mx) : 0.f;
    for (int m = t; m < Nm; m += 256) sum += expf(lrelu(sn[Nn + m] + sed) - mx);
    red[t] = sum; __syncthreads();
    for (int s = 128; s > 0; s >>= 1) { if (t < s) red[t] += red[t + s]; __syncthreads(); }
    float inv = 1.f / red[0];
    for (int m = t; m < Nm; m += 256)
        A[(size_t)d * Nm + m] = expf(lrelu(sn[Nn + m] + sed) - mx) * inv;
    if (t == 0) aself[d] = expf(vself - mx) * inv;
}

// --- E = Binv*(aself*Yself + Alpha@Ym); out_node = E*aself + b_gat
// --- one wave = 16 x 64 strip, 4 independent accumulator chains
__global__ void gemm_e_kernel(const float* __restrict__ Alpha, const float* __restrict__ Ym,
                              const float* __restrict__ Yself, const float* __restrict__ aself,
                              const float* __restrict__ bg, float* __restrict__ E,
                              float* __restrict__ outNode, float Binv, int Nm) {
    int tileM = blockIdx.x;
    int lane = threadIdx.x, half = lane >> 4, l16 = lane & 15;
    int rowA = tileM * 16 + l16;
    const float* Arow = Alpha + (size_t)rowA * Nm;
    v8f c0 = {}, c1 = {}, c2 = {}, c3 = {};
    for (int k0 = 0; k0 < Nm; k0 += 4) {
        int ka = k0 + 2 * half;
        v2f a;
        a.x = Arow[ka];
        a.y = Arow[ka + 1];
        const float* y0 = Ym + (size_t)ka * 64 + l16;
        const float* y1 = y0 + 64;
        v2f b0, b1, b2, b3;
        b0.x = y0[0];  b0.y = y1[0];
        b1.x = y0[16]; b1.y = y1[16];
        b2.x = y0[32]; b2.y = y1[32];
        b3.x = y0[48]; b3.y = y1[48];
        c0 = wmma_f32_k4(a, b0, c0);
        c1 = wmma_f32_k4(a, b1, c1);
        c2 = wmma_f32_k4(a, b2, c2);
        c3 = wmma_f32_k4(a, b3, c3);
    }
#pragma unroll
    for (int r = 0; r < 8; ++r) {
        int row = tileM * 16 + r + 8 * half;
        float as = aself[row];
        const float* ys = Yself + (size_t)row * 64 + l16;
        float e0 = (c0[r] + as * ys[0])  * Binv;
        float e1 = (c1[r] + as * ys[16]) * Binv;
        float e2 = (c2[r] + as * ys[32]) * Binv;
        float e3 = (c3[r] + as * ys[48]) * Binv;
        float* erow = E + (size_t)row * 64 + l16;
        erow[0]  = e0; erow[16] = e1; erow[32] = e2; erow[48] = e3;
        float* orow = outNode + (size_t)row * 128 + l16;   // Dinv = 1 for node rows
        orow[0]  = e0 * as + bg[l16];
        orow[16] = e1 * as + bg[l16 + 16];
        orow[32] = e2 * as + bg[l16 + 32];
        orow[48] = e3 * as + bg[l16 + 48];
    }
}

// ---------------- out_med = Dinv * Alpha^T @ E + b_gat   (transposed A read, 16x64 strip)
__global__ void gemm_t_kernel(const float* __restrict__ Alpha, const float* __restrict__ E,
                              const float* __restrict__ bg, float* __restrict__ outMed,
                              float Dinv, int Ne, int Nm) {
    int tileM = blockIdx.x;
    int lane = threadIdx.x, half = lane >> 4, l16 = lane & 15;
    int rowM = tileM * 16 + l16;          // medicine index
    v8f c0 = {}, c1 = {}, c2 = {}, c3 = {};
    for (int k0 = 0; k0 < Ne; k0 += 4) {
        int ka = k0 + 2 * half;
        v2f a;
        a.x = Alpha[(size_t)ka * Nm + rowM];
        a.y = Alpha[(size_t)(ka + 1) * Nm + rowM];
        const float* e0p = E + (size_t)ka * 64 + l16;
        const float* e1p = e0p + 64;
        v2f b0, b1, b2, b3;
        b0.x = e0p[0];  b0.y = e1p[0];
        b1.x = e0p[16]; b1.y = e1p[16];
        b2.x = e0p[32]; b2.y = e1p[32];
        b3.x = e0p[48]; b3.y = e1p[48];
        c0 = wmma_f32_k4(a, b0, c0);
        c1 = wmma_f32_k4(a, b1, c1);
        c2 = wmma_f32_k4(a, b2, c2);
        c3 = wmma_f32_k4(a, b3, c3);
    }
#pragma unroll
    for (int r = 0; r < 8; ++r) {
        int row = tileM * 16 + r + 8 * half;
        float* orow = outMed + (size_t)row * 128 + l16;
        orow[0]  = c0[r] * Dinv + bg[l16];
        orow[16] = c1[r] * Dinv + bg[l16 + 16];
        orow[32] = c2[r] * Dinv + bg[l16 + 32];
        orow[48] = c3[r] * Dinv + bg[l16 + 48];
    }
}

extern "C" void kernel_launch(void* const* d_in, const int* in_sizes, int n_in,
                              void* d_out, int out_size, void* d_ws, size_t ws_size,
                              hipStream_t stream) {
    (void)n_in; (void)out_size; (void)ws_size;
    const int*   diag_seq = (const int*)d_in[0];
    const int*   proc_seq = (const int*)d_in[1];
    const int*   med_seq  = (const int*)d_in[2];
    const float* c_emb    = (const float*)d_in[3];
    const float* p_emb    = (const float*)d_in[4];
    const float* m_emb    = (const float*)d_in[5];
    const float* W_conv   = (const float*)d_in[6];
    const float* b_conv   = (const float*)d_in[7];
    const float* W_gat    = (const float*)d_in[8];
    const float* b_gat    = (const float*)d_in[9];
    const float* att      = (const float*)d_in[10];
    const float* attr1    = (const float*)d_in[11];
    const float* attr2    = (const float*)d_in[12];
    const int Nd = in_sizes[0], Np = in_sizes[1], Nm = in_sizes[2];
    float* out = (float*)d_out;

    // workspace layout (floats)
    float* w = (float*)d_ws;
    size_t off = 0;
    float* Xd     = w + off; off += (size_t)Nd * 64;
    float* Xp     = w + off; off += (size_t)Np * 64;
    float* Xm     = w + off; off += (size_t)Nm * 64;
    float* colsum = w + off; off += 3 * 64;
    float* rows3  = w + off; off += 3 * 64;
    float* Y1     = w + off; off += (size_t)(Nd + Nm) * 64;
    float* Y2     = w + off; off += (size_t)(Np + Nm) * 64;
    float* He1    = w + off; off += (size_t)Nd * 64;
    float* He2    = w + off; off += (size_t)Np * 64;
    float* sn1    = w + off; off += (size_t)(Nd + Nm);
    float* se1    = w + off; off += (size_t)Nd;
    float* sn2    = w + off; off += (size_t)(Np + Nm);
    float* se2    = w + off; off += (size_t)Np;
    float* A1     = w + off; off += (size_t)Nd * Nm;
    float* as1    = w + off; off += (size_t)Nd;
    float* A2     = w + off; off += (size_t)Np * Nm;
    float* as2    = w + off; off += (size_t)Np;
    float* E1     = w + off; off += (size_t)Nd * 64;
    float* E2     = w + off; off += (size_t)Np * 64;

    // 1) gather embeddings
    gather_kernel<<<(Nd * 64 + 255) / 256, 256, 0, stream>>>(c_emb, diag_seq, Xd, Nd);
    gather_kernel<<<(Np * 64 + 255) / 256, 256, 0, stream>>>(p_emb, proc_seq, Xp, Np);
    gather_kernel<<<(Nm * 64 + 255) / 256, 256, 0, stream>>>(m_emb, med_seq, Xm, Nm);

    // 2) column sums -> mean rows through W_conv -> broadcast to out[:,64:128]
    colsum_kernel<<<64, 256, 0, stream>>>(Xd, colsum + 0,   Nd);
    colsum_kernel<<<64, 256, 0, stream>>>(Xp, colsum + 64,  Np);
    colsum_kernel<<<64, 256, 0, stream>>>(Xm, colsum + 128, Nm);
    meanrow_kernel<<<3, 64, 0, stream>>>(colsum, W_conv, b_conv, rows3, Nd, Np, Nm);
    broadcast_kernel<<<((Nd + Np + Nm) * 64 + 255) / 256, 256, 0, stream>>>(rows3, out, Nd, Np, Nm);

    // 3) X @ W_gat GEMMs (WMMA f32, 16x64 strip per wave): Y1=[Xd;Xm], Y2=[Xp;Xm], He1, He2
    gemm_xw_kernel<<<Nd / 16, 32, 0, stream>>>(Xd,    W_gat, Y1);
    gemm_xw_kernel<<<Nm / 16, 32, 0, stream>>>(Xm,    W_gat, Y1 + (size_t)Nd * 64);
    gemm_xw_kernel<<<Np / 16, 32, 0, stream>>>(Xp,    W_gat, Y2);
    gemm_xw_kernel<<<Nm / 16, 32, 0, stream>>>(Xm,    W_gat, Y2 + (size_t)Np * 64);
    gemm_xw_kernel<<<Nd / 16, 32, 0, stream>>>(attr1, W_gat, He1);
    gemm_xw_kernel<<<Np / 16, 32, 0, stream>>>(attr2, W_gat, He2);

    // 4) attention scores: s_node = Y @ att[:64], s_edge = He @ att[64:]
    rowdot_kernel<<<(Nd + Nm + 255) / 256, 256, 0, stream>>>(Y1,  att,      sn1, Nd + Nm);
    rowdot_kernel<<<(Nd + 255) / 256,      256, 0, stream>>>(He1, att + 64, se1, Nd);
    rowdot_kernel<<<(Np + Nm + 255) / 256, 256, 0, stream>>>(Y2,  att,      sn2, Np + Nm);
    rowdot_kernel<<<(Np + 255) / 256,      256, 0, stream>>>(He2, att + 64, se2, Np);

    // 5) per-edge softmax -> Alpha[Ne x Nm], alpha_self[Ne]
    softmax_kernel<<<Nd, 256, 0, stream>>>(sn1, se1, A1, as1, Nm, Nd);
    softmax_kernel<<<Np, 256, 0, stream>>>(sn2, se2, A2, as2, Nm, Np);

    // 6) E = Binv*(aself*Yself + Alpha@Ym), fused node-row output (WMMA)
    const float Binv = 1.0f / (float)(Nm + 1);
    gemm_e_kernel<<<Nd / 16, 32, 0, stream>>>(A1, Y1 + (size_t)Nd * 64, Y1, as1, b_gat,
                                              E1, out, Binv, Nm);
    gemm_e_kernel<<<Np / 16, 32, 0, stream>>>(A2, Y2 + (size_t)Np * 64, Y2, as2, b_gat,
                                              E2, out + (size_t)Nd * 128, Binv, Nm);

    // 7) medicine rows: (1/Nd) * A1^T @ E1 + b_gat  (pro_med medicine part unused)
    gemm_t_kernel<<<Nm / 16, 32, 0, stream>>>(A1, E1, b_gat,
                                              out + (size_t)(Nd + Np) * 128,
                                              1.0f / (float)Nd, Nd, Nm);
}